// SerializationLayer_33689723470058
// MI455X (gfx1250) — compile-verified
//
#include <hip/hip_runtime.h>
#include <stdint.h>

// ---------------------------------------------------------------------------
// SerializationLayer for MI455X (gfx1250).
// Bandwidth-bound: ~142 MB stores + 75 MB coord loads => ~9.3 us floor at
// 23.3 TB/s. Strategy: single pass over coords, 128-bit coalesced accesses,
// async global->LDS staging (GLOBAL_LOAD_ASYNC_TO_LDS_B128 + s_wait_asynccnt)
// for the coords stream, non-temporal (TH=NT) stores for the write-once
// output streams, binary-search instead of a full atomic counting pass
// (batch ids are sorted by construction).
// ---------------------------------------------------------------------------

#define THREADS 256
#define PTS_PER_THREAD 4
#define TILE_PTS (THREADS * PTS_PER_THREAD)   // 1024 points per block
#define TILE_INT4 (TILE_PTS * 3 / 4)          // 768 int4 = 12288 bytes

#if defined(__gfx1250__) && __has_builtin(__builtin_amdgcn_global_load_async_to_lds_b128)
#define USE_ASYNC_LDS 1
#else
#define USE_ASYNC_LDS 0
#endif

#if USE_ASYNC_LDS
typedef int v4i __attribute__((ext_vector_type(4)));
typedef __attribute__((address_space(1))) v4i* glb_v4i_p;
typedef __attribute__((address_space(3))) v4i* lds_v4i_p;
#endif

typedef float v4f __attribute__((ext_vector_type(4)));

__device__ __forceinline__ void store_nt4(float* p, float a, float b, float c,
                                          float d) {
    v4f v = {a, b, c, d};
    __builtin_nontemporal_store(v, (v4f*)p);   // global_store_b128 th:NT
}

__device__ __forceinline__ unsigned part1by1(unsigned v) {
    v &= 0xFFFFu;
    v = (v | (v << 8)) & 0x00FF00FFu;
    v = (v | (v << 4)) & 0x0F0F0F0Fu;
    v = (v | (v << 2)) & 0x33333333u;
    v = (v | (v << 1)) & 0x55555555u;
    return v;
}

// ---------------------------------------------------------------------------
// Kernel 1: batch ids in coords[:,0] are sorted (repeat(arange(bs))), so find
// segment starts with per-thread binary search instead of a 75MB counting
// pass. starts[bs] = n_total.
// ---------------------------------------------------------------------------
__global__ void find_starts_kernel(const int* __restrict__ coords,
                                   const int* __restrict__ batch_size_p,
                                   int n_total,
                                   int* __restrict__ starts) {
    const int bs = *batch_size_p;
    const int t = (int)threadIdx.x;
    if (t > bs) return;
    if (t == bs) { starts[t] = n_total; return; }
    // lower_bound of value t over coords[:,0]
    int lo = 0, hi = n_total;
    while (lo < hi) {
        int mid = (lo + hi) >> 1;
        if (coords[(size_t)mid * 3] < t) lo = mid + 1; else hi = mid;
    }
    starts[t] = lo;
}

// ---------------------------------------------------------------------------
// Kernel 2: flat2win + mask over the padded index space. Pure index math,
// two float4 NT stores per thread. max_voxels % 4 == 0 keeps each thread's 4
// consecutive j inside one batch.
// ---------------------------------------------------------------------------
__global__ void padded_kernel(const int* __restrict__ starts,
                              const int* __restrict__ max_voxels_p,
                              float* __restrict__ out_flat2win,
                              float* __restrict__ out_mask,
                              unsigned total_padded) {
    const unsigned j0 = (blockIdx.x * blockDim.x + threadIdx.x) * 4u;
    if (j0 >= total_padded) return;
    const unsigned mv = (unsigned)*max_voxels_p;
    const unsigned i = j0 / mv;
    const unsigned local0 = j0 - i * mv;
    const int st = starts[i];
    const int nr = starts[i + 1] - st;
    int period = 2 * nr - 2;
    if (period < 1) period = 1;

    float f[4], m[4];
#pragma unroll
    for (int r = 0; r < 4; ++r) {
        const int local = (int)(local0 + (unsigned)r);
        const bool msk = local >= nr;
        int f2w;
        if (!msk) {
            f2w = st + local;
        } else {
            const int k = (local - nr) % period;           // non-negative
            const int mirror = (k < nr - 1) ? (nr - 2 - k) : (k - nr + 2);
            f2w = st + mirror;
        }
        f[r] = (float)f2w;
        m[r] = msk ? 1.0f : 0.0f;
    }
    store_nt4(out_flat2win + j0, f[0], f[1], f[2], f[3]);
    store_nt4(out_mask + j0,     m[0], m[1], m[2], m[3]);
}

// ---------------------------------------------------------------------------
// Kernel 3: win2flat + Morton codes. Coords tile (12KB) staged to LDS via
// wave-private async global->LDS b128 transfers (3 per lane), then read back
// as aligned ds_load_b128 (stride-3 AoS = 48B/thread). Three float4 NT stores.
// ---------------------------------------------------------------------------
__global__ void real_kernel(const int* __restrict__ coords,
                            const int* __restrict__ starts,
                            const int* __restrict__ max_voxels_p,
                            float* __restrict__ out_win2flat,
                            float* __restrict__ out_z,
                            float* __restrict__ out_zt,
                            int n_total) {
#if USE_ASYNC_LDS
    __shared__ int4 tile[TILE_INT4];   // 12288 bytes
#endif
    const int t = (int)threadIdx.x;
    const int tile_base_pt = (int)blockIdx.x * TILE_PTS;
    const bool full = (tile_base_pt + TILE_PTS) <= n_total;
    const int mv = *max_voxels_p;

    int c[12];
    if (full) {
#if USE_ASYNC_LDS
        const int wave = t >> 5;
        const int lane = t & 31;
        const char* gbase = (const char*)coords + (size_t)tile_base_pt * 12u;
        // Wave w copies byte range [w*1536, (w+1)*1536): 32 lanes x 16B x 3.
        // Wave-private staging: each wave reads back only what it wrote, so
        // s_wait_asynccnt 0 suffices (no workgroup barrier needed).
#pragma unroll
        for (int k = 0; k < 3; ++k) {
            const int off = wave * 1536 + lane * 16 + k * 512;
            __builtin_amdgcn_global_load_async_to_lds_b128(
                (glb_v4i_p)(gbase + off),
                (lds_v4i_p)((char*)tile + off),
                0, 0);
        }
#if __has_builtin(__builtin_amdgcn_s_wait_asynccnt)
        __builtin_amdgcn_s_wait_asynccnt(0);
#else
        asm volatile("s_wait_asynccnt 0" ::: "memory");
#endif
        const int4 a  = tile[t * 3 + 0];
        const int4 b4 = tile[t * 3 + 1];
        const int4 cc = tile[t * 3 + 2];
#else
        const int4* g4 = (const int4*)(coords + (size_t)tile_base_pt * 3u);
        const int4 a  = g4[t * 3 + 0];
        const int4 b4 = g4[t * 3 + 1];
        const int4 cc = g4[t * 3 + 2];
#endif
        c[0] = a.x;  c[1] = a.y;  c[2]  = a.z;  c[3]  = a.w;
        c[4] = b4.x; c[5] = b4.y; c[6]  = b4.z; c[7]  = b4.w;
        c[8] = cc.x; c[9] = cc.y; c[10] = cc.z; c[11] = cc.w;
    } else {
#pragma unroll
        for (int p = 0; p < 4; ++p) {
            const int idx = tile_base_pt + t * 4 + p;
            if (idx < n_total) {
                c[p * 3 + 0] = coords[(size_t)idx * 3 + 0];
                c[p * 3 + 1] = coords[(size_t)idx * 3 + 1];
                c[p * 3 + 2] = coords[(size_t)idx * 3 + 2];
            } else {
                c[p * 3 + 0] = 0; c[p * 3 + 1] = 0; c[p * 3 + 2] = 0;
            }
        }
    }

    float w2f[4], zc[4], zt[4];
#pragma unroll
    for (int p = 0; p < 4; ++p) {
        const int idx = tile_base_pt + t * 4 + p;
        const int b = c[p * 3 + 0];
        const unsigned y = (unsigned)c[p * 3 + 1];
        const unsigned x = (unsigned)c[p * 3 + 2];
        const int st = starts[b];
        const int wf = idx - st + b * mv;
        const unsigned px = part1by1(x);
        const unsigned py = part1by1(y);
        const unsigned z   = (px | (py << 1)) + ((unsigned)b << 16);  // 2*DEPTH=16
        const unsigned ztc = (py | (px << 1)) + ((unsigned)b << 16);
        w2f[p] = (float)wf;
        zc[p]  = (float)z;
        zt[p]  = (float)ztc;
    }

    const int base = tile_base_pt + t * 4;
    if (full) {
        store_nt4(out_win2flat + base, w2f[0], w2f[1], w2f[2], w2f[3]);
        store_nt4(out_z + base,        zc[0], zc[1], zc[2], zc[3]);
        store_nt4(out_zt + base,       zt[0], zt[1], zt[2], zt[3]);
    } else {
#pragma unroll
        for (int p = 0; p < 4; ++p) {
            const int idx = base + p;
            if (idx < n_total) {
                out_win2flat[idx] = w2f[p];
                out_z[idx]        = zc[p];
                out_zt[idx]       = zt[p];
            }
        }
    }
}

// ---------------------------------------------------------------------------
// Launch. Inputs (setup_inputs order): coords (int, n_total*3), batch_size
// (int scalar on device), max_voxels (int scalar on device), sparse_shape
// (unused). Output layout (float, concatenated in return order):
//   [0,P) flat2win | [P,P+N) win2flat | [P+N,2P+N) mask |
//   [2P+N,2P+2N) z_code | [2P+2N,2P+3N) z_trans_code
// where N = n_total, P = total padded = (out_size - 3N)/2.
// ---------------------------------------------------------------------------
extern "C" void kernel_launch(void* const* d_in, const int* in_sizes, int n_in,
                              void* d_out, int out_size, void* d_ws, size_t ws_size,
                              hipStream_t stream) {
    const int* coords = (const int*)d_in[0];
    const int* bs_p   = (const int*)d_in[1];
    const int* mv_p   = (const int*)d_in[2];

    const int n_total = in_sizes[0] / 3;
    const long long P = ((long long)out_size - 3LL * (long long)n_total) / 2LL;
    const unsigned total_padded = (unsigned)P;

    float* out          = (float*)d_out;
    float* out_flat2win = out;
    float* out_win2flat = out + P;
    float* out_mask     = out + P + n_total;
    float* out_z        = out + 2 * P + n_total;
    float* out_zt       = out + 2 * P + 2 * (long long)n_total;

    int* starts = (int*)d_ws;  // batch_size+1 ints (<=64B), written each call

    find_starts_kernel<<<1, 256, 0, stream>>>(coords, bs_p, n_total, starts);

    const unsigned pblocks = (total_padded + (THREADS * 4u) - 1u) / (THREADS * 4u);
    padded_kernel<<<pblocks, THREADS, 0, stream>>>(starts, mv_p, out_flat2win,
                                                   out_mask, total_padded);

    const int rblocks = (n_total + TILE_PTS - 1) / TILE_PTS;
    real_kernel<<<rblocks, THREADS, 0, stream>>>(coords, starts, mv_p,
                                                 out_win2flat, out_z, out_zt,
                                                 n_total);
}